// MatrixPowerSeriesLayer_3925600108784
// MI455X (gfx1250) — compile-verified
//
#include <hip/hip_runtime.h>

typedef __attribute__((ext_vector_type(2))) float v2f;
typedef __attribute__((ext_vector_type(8))) float v8f;

#define MATN     128
#define STRIDE   132                       // padded row stride (floats) -> conflict-free DS
#define XT_SZ    (MATN * STRIDE)           // 16896 floats per component of X^T
#define ROWS     64                        // rows handled per block (4 waves x 16)
#define TBUF_C   (ROWS * STRIDE)           // 8448 floats, one component of T strip
#define TBUF_SZ  (2 * TBUF_C)              // re+im, one buffer
#define T_OFF    (2 * XT_SZ)               // T buffers start after Xt_re, Xt_im
#define LDS_FLT  (T_OFF + 2 * TBUF_SZ)     // 67584 floats
#define LDS_BYTES (LDS_FLT * 4)            // 270336 bytes (< 320KB per WGP)
#define SLEN     10

__launch_bounds__(128, 1)
__global__ void mps_kernel(const float* __restrict__ x,
                           const float* __restrict__ coef,
                           float* __restrict__ out)
{
    extern __shared__ float sm[];
    const int b    = blockIdx.x >> 1;
    const int rowB = (blockIdx.x & 1) * ROWS;    // global row base of this block's strip
    const int tid  = threadIdx.x;                // 0..127
    const int wave = tid >> 5;                   // 0..3
    const int lane = tid & 31;

    const float* xr = x + (size_t)b * 2 * MATN * MATN;
    const float* xi = xr + MATN * MATN;

    // ---- Cooperative LDS init: X^T (full matrix, both components) + T buffer 0 (block strip)
    for (int it = 0; it < 32; ++it) {
        int idx4 = (it * 128 + tid) * 4;         // element index, 4 at a time
        int r = idx4 >> 7;                       // row 0..127
        int c = idx4 & 127;                      // col, multiple of 4
        float4 vr = *(const float4*)(xr + idx4);
        float4 vi = *(const float4*)(xi + idx4);
        // transposed: Xt[c][r]
        sm[(c + 0) * STRIDE + r] = vr.x;
        sm[(c + 1) * STRIDE + r] = vr.y;
        sm[(c + 2) * STRIDE + r] = vr.z;
        sm[(c + 3) * STRIDE + r] = vr.w;
        sm[XT_SZ + (c + 0) * STRIDE + r] = vi.x;
        sm[XT_SZ + (c + 1) * STRIDE + r] = vi.y;
        sm[XT_SZ + (c + 2) * STRIDE + r] = vi.z;
        sm[XT_SZ + (c + 3) * STRIDE + r] = vi.w;
        // row-major T strip, buffer 0 (T_1 = X)
        if (r >= rowB && r < rowB + ROWS) {
            int rl = r - rowB;
            *(float4*)&sm[T_OFF + 0 * TBUF_C + rl * STRIDE + c] = vr;
            *(float4*)&sm[T_OFF + 1 * TBUF_C + rl * STRIDE + c] = vi;
        }
    }
    __syncthreads();

    // ---- WMMA fragment lane decomposition (16x16 tiles, wave32)
    const int lhalf = lane >> 4;                 // 0: lanes 0-15, 1: lanes 16-31
    const int l15   = lane & 15;
    const int lr0   = wave * 16;                 // local row base of this wave's strip

    // ---- Series accumulators R = c0*I + c1*X  (C/D layout: VGPR v -> M = v + 8*lhalf, N = l15)
    const float c0r = coef[0], c0i = coef[1], c1r = coef[2], c1i = coef[3];
    v8f Rr[8], Ri[8];
    #pragma unroll
    for (int j = 0; j < 8; ++j) {
        int cg = j * 16 + l15;
        #pragma unroll
        for (int v = 0; v < 8; ++v) {
            int rg = rowB + lr0 + v + 8 * lhalf;
            float xre = sm[cg * STRIDE + rg];            // Xt[cg][rg] == X[rg][cg]
            float xim = sm[XT_SZ + cg * STRIDE + rg];
            float idv = (rg == cg) ? 1.0f : 0.0f;
            Rr[j][v] = c0r * idv + c1r * xre - c1i * xim;
            Ri[j][v] = c0i * idv + c1i * xre + c1r * xim;
        }
    }

    // ---- Power iteration: T_k = T_{k-1} * X (complex), R += c_k * T_k
    int p = 0;
    #pragma clang loop unroll(disable)
    for (int step = 2; step < SLEN; ++step) {
        const float cr = coef[2 * step + 0];
        const float ci = coef[2 * step + 1];
        const float* Tre = &sm[T_OFF + p * TBUF_SZ];
        const float* Tim = Tre + TBUF_C;
        float* Ure = &sm[T_OFF + (1 - p) * TBUF_SZ];
        float* Uim = Ure + TBUF_C;

        // A fragment base: row = lr0 + l15, K pair selected by lane half.
        // Hoist the wave's entire A strip (invariant across j) into registers:
        // 32 k-tiles x 2 components x v2f = 128 VGPRs.
        const int arow = (lr0 + l15) * STRIDE + 2 * lhalf;
        const float* Abre = Tre + arow;
        const float* Abim = Tim + arow;
        v2f Are[32], Aim[32];
        #pragma unroll
        for (int k = 0; k < 32; ++k) {
            Are[k] = *(const v2f*)(Abre + 4 * k);
            Aim[k] = *(const v2f*)(Abim + 4 * k);
        }

        #pragma unroll
        for (int j = 0; j < 8; ++j) {
            const int bcol = (j * 16 + l15) * STRIDE + 2 * lhalf;
            const float* Bre = sm + bcol;
            const float* Bim = sm + XT_SZ + bcol;
            // Four balanced, independent accumulator chains (32 WMMAs each) so the
            // scheduler can interleave them and hide XDL accumulation latency.
            v8f P = {}, Q = {}, S1 = {}, S2 = {};
            #pragma unroll
            for (int k = 0; k < 32; ++k) {       // K = 128 in chunks of 4
                v2f bre = *(const v2f*)(Bre + 4 * k);
                v2f bim = *(const v2f*)(Bim + 4 * k);
                P  = __builtin_amdgcn_wmma_f32_16x16x4_f32(false, Are[k], false, bre, (short)0, P,  false, false);
                Q  = __builtin_amdgcn_wmma_f32_16x16x4_f32(false, Aim[k], false, bim, (short)0, Q,  false, false);
                S1 = __builtin_amdgcn_wmma_f32_16x16x4_f32(false, Are[k], false, bim, (short)0, S1, false, false);
                S2 = __builtin_amdgcn_wmma_f32_16x16x4_f32(false, Aim[k], false, bre, (short)0, S2, false, false);
            }
            // new_tr = P - Q, new_ti = S1 + S2 ; accumulate series and stage new T tile
            #pragma unroll
            for (int v = 0; v < 8; ++v) {
                float ntr = P[v] - Q[v];
                float nti = S1[v] + S2[v];
                Rr[j][v] += cr * ntr - ci * nti;
                Ri[j][v] += ci * ntr + cr * nti;
                int rl = lr0 + v + 8 * lhalf;
                Ure[rl * STRIDE + j * 16 + l15] = ntr;
                Uim[rl * STRIDE + j * 16 + l15] = nti;
            }
        }
        p ^= 1;
    }

    // ---- Write result strip: out[b][0][r][c], out[b][1][r][c]
    float* outp = out + (size_t)b * 2 * MATN * MATN;
    #pragma unroll
    for (int j = 0; j < 8; ++j) {
        int cg = j * 16 + l15;
        #pragma unroll
        for (int v = 0; v < 8; ++v) {
            int rg = rowB + lr0 + v + 8 * lhalf;
            outp[rg * MATN + cg]               = Rr[j][v];
            outp[MATN * MATN + rg * MATN + cg] = Ri[j][v];
        }
    }
}

extern "C" void kernel_launch(void* const* d_in, const int* in_sizes, int n_in,
                              void* d_out, int out_size, void* d_ws, size_t ws_size,
                              hipStream_t stream) {
    const float* x    = (const float*)d_in[0];   // (512, 2, 128, 128) fp32
    const float* coef = (const float*)d_in[1];   // (10, 2) fp32
    float* out        = (float*)d_out;           // (512, 2, 128, 128) fp32

    (void)in_sizes; (void)n_in; (void)out_size; (void)d_ws; (void)ws_size;

    // Opt in to >64KB dynamic LDS (deterministic, not a stream op; capture-safe).
    hipFuncSetAttribute((const void*)mps_kernel,
                        hipFuncAttributeMaxDynamicSharedMemorySize, LDS_BYTES);

    // 512 batches x 2 row-strips of 64 rows; 128 threads = 4 waves (16 rows each)
    mps_kernel<<<dim3(1024), dim3(128), LDS_BYTES, stream>>>(x, coef, out);
}